// hybrid_4415226380888
// MI455X (gfx1250) — compile-verified
//
#include <hip/hip_runtime.h>
#include <cstdint>

// ---------------- problem constants ----------------
#define S_DIM   90
#define B_DIM   64
#define V_DIM   32000
#define N_ROWS  (S_DIM * B_DIM)        // 5760
#define KWIN    30
#define ALPHA_F 0.7f
#define TPB     256
#define L2E     1.44269504088896340736f

// ---------------- chunking for LDS staging ----------------
// 1792 float4 = 7168 floats = 28 KB per buffer; double buffered = 56 KB LDS.
// 1792 / 256 threads = exactly 7 async-load instructions per wave per chunk,
// so s_wait_asynccnt immediates are compile-time constants.
#define CHUNK_F4     1792
#define CHUNK_FLOATS (CHUNK_F4 * 4)    // 7168
#define NFULL        4                 // 4 * 7168 = 28672 staged via async
#define TAIL_START   (NFULL * CHUNK_FLOATS)        // 28672
#define TAIL_F4      ((V_DIM - TAIL_START) / 4)    // 832 (direct global loads)

// ---------------- CDNA5 async global->LDS helpers ----------------
typedef int vint4 __attribute__((vector_size(16)));

__device__ __forceinline__ void async_load_f4(const float4* g, float4* l) {
#if __has_builtin(__builtin_amdgcn_global_load_async_to_lds_b128)
  __builtin_amdgcn_global_load_async_to_lds_b128(
      (__attribute__((address_space(1))) vint4*)(uintptr_t)g,
      (__attribute__((address_space(3))) vint4*)(uint32_t)(uintptr_t)l,
      0, 0);
#else
  *l = *g;  // synchronous fallback; waits below become no-ops
#endif
}

#if __has_builtin(__builtin_amdgcn_s_wait_asynccnt)
#define WAIT_ASYNC(n) __builtin_amdgcn_s_wait_asynccnt(n)
#else
#define WAIT_ASYNC(n) asm volatile("s_wait_asynccnt %0" ::"n"(n) : "memory")
#endif

// ---------------- row softmax statistics (shared impl) ----------------
// One block per row of [*, V_DIM] f32. Online (max, sum-exp) with 4
// independent accumulator states (one per float4 lane) for ILP, then a
// deterministic 256-thread LDS tree combine.
// ARGMAX=false (scores): writes ce_num[n] = nll*w, ce_den[n] = w.
// ARGMAX=true  (pred)  : writes rowmax[n], rowZ[n] (= sum e^{x-max}), predidx[n].
template <bool ARGMAX>
__device__ __forceinline__ void row_softmax_impl(
    const float* __restrict__ X, const int* __restrict__ gtruth,
    const float* __restrict__ weight, float* __restrict__ out_a,
    float* __restrict__ out_b, int* __restrict__ out_idx) {
  __shared__ float4 buf[2][CHUNK_F4];
  __shared__ float  red_m[TPB];
  __shared__ float  red_s[TPB];
  __shared__ int    red_i[TPB];

  const int n   = blockIdx.x;
  const int tid = threadIdx.x;
  const float4* __restrict__ grow = (const float4*)(X + (size_t)n * V_DIM);

  const float NEG_INF = -__builtin_inff();
  float m0 = NEG_INF, m1 = NEG_INF, m2 = NEG_INF, m3 = NEG_INF;
  float s0 = 0.f, s1 = 0.f, s2 = 0.f, s3 = 0.f;
  int   b0 = 0, b1 = 0, b2 = 0, b3 = 0;

#define UPD(Mk, Sk, Bk, x, gi)                              \
  do {                                                      \
    float _x = (x);                                         \
    if (_x > Mk) {                                          \
      Sk = Sk * exp2f((Mk - _x) * L2E) + 1.0f;              \
      Mk = _x;                                              \
      if (ARGMAX) Bk = (gi);                                \
    } else {                                                \
      Sk += exp2f((_x - Mk) * L2E);                         \
    }                                                       \
  } while (0)

  // Prime the double buffer: issue chunks 0 and 1 (7 async instrs/wave each).
  for (int j = tid; j < CHUNK_F4; j += TPB) async_load_f4(grow + j, &buf[0][j]);
  for (int j = tid; j < CHUNK_F4; j += TPB)
    async_load_f4(grow + CHUNK_F4 + j, &buf[1][j]);

  for (int c = 0; c < NFULL; ++c) {
    // Outstanding beyond chunk c is exactly one chunk (7 instrs) except last.
    if (c < NFULL - 1) { WAIT_ASYNC(7); } else { WAIT_ASYNC(0); }
    __syncthreads();
    const int base = c * CHUNK_FLOATS;
    for (int j = tid; j < CHUNK_F4; j += TPB) {
      float4 v = buf[c & 1][j];
      int gi   = base + 4 * j;
      UPD(m0, s0, b0, v.x, gi);
      UPD(m1, s1, b1, v.y, gi + 1);
      UPD(m2, s2, b2, v.z, gi + 2);
      UPD(m3, s3, b3, v.w, gi + 3);
    }
    __syncthreads();  // everyone done reading buf[c&1] before overwrite
    if (c + 2 < NFULL) {
      for (int j = tid; j < CHUNK_F4; j += TPB)
        async_load_f4(grow + (size_t)(c + 2) * CHUNK_F4 + j, &buf[c & 1][j]);
    }
  }

  // Tail (3328 floats): direct global float4 loads.
  for (int j = tid; j < TAIL_F4; j += TPB) {
    float4 v = grow[TAIL_START / 4 + j];
    int gi   = TAIL_START + 4 * j;
    UPD(m0, s0, b0, v.x, gi);
    UPD(m1, s1, b1, v.y, gi + 1);
    UPD(m2, s2, b2, v.z, gi + 2);
    UPD(m3, s3, b3, v.w, gi + 3);
  }
#undef UPD

  // Merge the 4 per-thread states (exact tie -> smaller index).
  auto merge = [&](float& ma, float& sa, int& ia, float mb, float sb, int ib) {
    float M = fmaxf(ma, mb);
    sa      = sa * exp2f((ma - M) * L2E) + sb * exp2f((mb - M) * L2E);
    ia      = (ma > mb) ? ia : ((mb > ma) ? ib : min(ia, ib));
    ma      = M;
  };
  merge(m0, s0, b0, m1, s1, b1);
  merge(m2, s2, b2, m3, s3, b3);
  merge(m0, s0, b0, m2, s2, b2);

  // Deterministic tree combine across 256 threads.
  red_m[tid] = m0; red_s[tid] = s0; red_i[tid] = b0;
  __syncthreads();
  for (int off = TPB / 2; off > 0; off >>= 1) {
    if (tid < off) {
      float ma = red_m[tid], mb = red_m[tid + off];
      float sa = red_s[tid], sb = red_s[tid + off];
      int   ia = red_i[tid], ib = red_i[tid + off];
      float M  = fmaxf(ma, mb);
      red_s[tid] = sa * exp2f((ma - M) * L2E) + sb * exp2f((mb - M) * L2E);
      red_m[tid] = M;
      red_i[tid] = (ma > mb) ? ia : ((mb > ma) ? ib : min(ia, ib));
    }
    __syncthreads();
  }

  if (tid == 0) {
    if (ARGMAX) {
      out_a[n]   = red_m[0];
      out_b[n]   = red_s[0];
      out_idx[n] = red_i[0];
    } else {
      int   g   = gtruth[n];
      float xt  = X[(size_t)n * V_DIM + g];
      float nll = red_m[0] + logf(red_s[0]) - xt;
      float w   = (g != 0) ? weight[g] : 0.0f;
      out_a[n]  = nll * w;
      out_b[n]  = w;
    }
  }
}

// Concrete (non-template) kernels so they appear first in the emitted asm.
__global__ __launch_bounds__(TPB) void scores_row_kernel(
    const float* __restrict__ scores, const int* __restrict__ gtruth,
    const float* __restrict__ weight, float* __restrict__ ce_num,
    float* __restrict__ ce_den) {
  row_softmax_impl<false>(scores, gtruth, weight, ce_num, ce_den, nullptr);
}

__global__ __launch_bounds__(TPB) void pred_row_kernel(
    const float* __restrict__ pred, float* __restrict__ rowmax,
    float* __restrict__ rowZ, int* __restrict__ predidx) {
  row_softmax_impl<true>(pred, nullptr, nullptr, rowmax, rowZ, predidx);
}

// ---------------- tu: unique-token ratio per batch column ----------------
__global__ __launch_bounds__(TPB) void tu_kernel(
    const int* __restrict__ predidx, float* __restrict__ tu_part) {
  __shared__ unsigned bits[V_DIM / 32];  // 32000-bit set
  __shared__ int      red[TPB];
  __shared__ int      wc_sh;
  const int b = blockIdx.x, tid = threadIdx.x;
  for (int j = tid; j < V_DIM / 32; j += TPB) bits[j] = 0u;
  if (tid == 0) wc_sh = 0;
  __syncthreads();
  int wc = 0;
  for (int sidx = tid; sidx < S_DIM; sidx += TPB) {
    int t = predidx[sidx * B_DIM + b];
    if (t != 0) {
      wc++;
      atomicOr(&bits[t >> 5], 1u << (t & 31));
    }
  }
  atomicAdd(&wc_sh, wc);
  __syncthreads();
  int uq = 0;
  for (int j = tid; j < V_DIM / 32; j += TPB) uq += __popc(bits[j]);
  red[tid] = uq;
  __syncthreads();
  for (int off = TPB / 2; off > 0; off >>= 1) {
    if (tid < off) red[tid] += red[tid + off];
    __syncthreads();
  }
  if (tid == 0) tu_part[b] = (float)red[0] / (float)wc_sh;
}

// ---------------- windowed duplicate penalty ----------------
__global__ __launch_bounds__(TPB) void window_kernel(
    const float* __restrict__ pred, const float* __restrict__ rowmax,
    const float* __restrict__ rowZ, const int* __restrict__ predidx,
    float* __restrict__ wsum) {
  __shared__ float m_sh[B_DIM], zi_sh[B_DIM];
  __shared__ float red[TPB];
  const int i = blockIdx.x, tid = threadIdx.x;
  if (tid < B_DIM) {
    int r      = i * B_DIM + tid;
    m_sh[tid]  = rowmax[r];
    zi_sh[tid] = 1.0f / rowZ[r];
  }
  __syncthreads();
  int lo = i - KWIN; if (lo < 0) lo = 0;
  const int total = (i - lo) * B_DIM;  // <= 1920 gathers
  float acc = 0.0f;
  for (int p = tid; p < total; p += TPB) {
    int sp = lo + (p >> 6);
    int b  = p & (B_DIM - 1);
    int t  = predidx[sp * B_DIM + b];
    float x = pred[(size_t)(i * B_DIM + b) * V_DIM + t];
    acc += exp2f((x - m_sh[b]) * L2E) * zi_sh[b];
  }
  red[tid] = acc;
  __syncthreads();
  for (int off = TPB / 2; off > 0; off >>= 1) {
    if (tid < off) red[tid] += red[tid + off];
    __syncthreads();
  }
  if (tid == 0) {
    int d = (i < KWIN) ? i : KWIN;
    if (d < 1) d = 1;
    wsum[i] = red[0] / (float)d;
  }
}

// ---------------- final scalar combine ----------------
__global__ __launch_bounds__(TPB) void finalize_kernel(
    const float* __restrict__ ce_num, const float* __restrict__ ce_den,
    const float* __restrict__ wsum, const float* __restrict__ tu_part,
    float* __restrict__ out) {
  __shared__ float red[TPB];
  const int tid = threadIdx.x;
  auto block_sum = [&](float v) -> float {
    red[tid] = v;
    __syncthreads();
    for (int off = TPB / 2; off > 0; off >>= 1) {
      if (tid < off) red[tid] += red[tid + off];
      __syncthreads();
    }
    float r = red[0];
    __syncthreads();
    return r;
  };
  float a = 0.f, b = 0.f, c = 0.f, d = 0.f;
  for (int j = tid; j < N_ROWS; j += TPB) { a += ce_num[j]; b += ce_den[j]; }
  for (int j = tid; j < S_DIM; j += TPB) c += wsum[j];
  for (int j = tid; j < B_DIM; j += TPB) d += tu_part[j];
  float num = block_sum(a);
  float den = block_sum(b);
  float summ = block_sum(c);
  float tusum = block_sum(d);
  if (tid == 0) {
    float cross = num / den;
    float tu    = tusum / (float)B_DIM;
    out[0] = (1.0f - ALPHA_F) * cross + ALPHA_F * summ;  // totalloss
    out[1] = cross;
    out[2] = tu;
  }
}

// ---------------- host launcher ----------------
extern "C" void kernel_launch(void* const* d_in, const int* in_sizes, int n_in,
                              void* d_out, int out_size, void* d_ws,
                              size_t ws_size, hipStream_t stream) {
  (void)in_sizes; (void)n_in; (void)out_size; (void)ws_size;
  const float* scores = (const float*)d_in[0];
  const int*   gtruth = (const int*)d_in[1];
  const float* pred   = (const float*)d_in[2];
  const float* weight = (const float*)d_in[3];
  float* out = (float*)d_out;

  float* ws      = (float*)d_ws;
  float* ce_num  = ws;                    // N_ROWS
  float* ce_den  = ws + N_ROWS;           // N_ROWS
  float* rowmax  = ws + 2 * N_ROWS;       // N_ROWS
  float* rowZ    = ws + 3 * N_ROWS;       // N_ROWS
  float* wsum    = ws + 4 * N_ROWS;       // S_DIM
  float* tu_part = wsum + S_DIM;          // B_DIM
  int*   predidx = (int*)(tu_part + B_DIM);  // N_ROWS ints

  scores_row_kernel<<<N_ROWS, TPB, 0, stream>>>(scores, gtruth, weight, ce_num,
                                                ce_den);
  pred_row_kernel<<<N_ROWS, TPB, 0, stream>>>(pred, rowmax, rowZ, predidx);
  tu_kernel<<<B_DIM, TPB, 0, stream>>>(predidx, tu_part);
  window_kernel<<<S_DIM, TPB, 0, stream>>>(pred, rowmax, rowZ, predidx, wsum);
  finalize_kernel<<<1, TPB, 0, stream>>>(ce_num, ce_den, wsum, tu_part, out);
}